// DecoderBlock_26431228739714
// MI455X (gfx1250) — compile-verified
//
#include <hip/hip_runtime.h>
#include <math.h>

// CDNA5 / gfx1250, wave32.
typedef __attribute__((ext_vector_type(2))) float v2f;
typedef __attribute__((ext_vector_type(8))) float v8f;

#define HID 128
#define NHEAD 8

// Padded LDS strides (floats). 64 banks x 4B:
//  A stride 132: bank(mrow*132+k) = (4*mrow + k)%64 -> 16 lanes hit 16 banks;
//                lanes 16-31 read k+2 -> disjoint bank set. Conflict-free.
//  W stride 520/392: 2*stride % 64 == 16 -> the two lane-halves (rows k, k+2)
//                land in disjoint 16-bank groups. Conflict-free.
#define LDA 132
#define LDW_QKV 520
#define LDW_VEC 392

// ---------------------------------------------------------------------------
// async global->LDS staging (CDNA5 GLOBAL_LOAD_ASYNC_TO_LDS_B128, ASYNCcnt),
// with a portable fallback if the builtins are not exposed.
// Probe feedback (round 2): param 1 is `__attribute__((vector_size(16))) int *`.
// ---------------------------------------------------------------------------
#if defined(__has_builtin)
#  if __has_builtin(__builtin_amdgcn_global_load_async_to_lds_b128) && \
      __has_builtin(__builtin_amdgcn_s_wait_asynccnt)
#    define USE_ASYNC 1
#  endif
#endif
#ifndef USE_ASYNC
#  define USE_ASYNC 0
#endif

typedef int v4i __attribute__((vector_size(16)));
typedef __attribute__((address_space(1))) v4i gv4i;   // global
typedef __attribute__((address_space(3))) v4i lv4i;   // LDS

__device__ __forceinline__ void cp_b128(const float* gsrc, float* ldst) {
#if USE_ASYNC
    __builtin_amdgcn_global_load_async_to_lds_b128(
        (gv4i*)gsrc, (lv4i*)ldst, 0, 0);
#else
    *(float4*)ldst = *(const float4*)gsrc;
#endif
}

__device__ __forceinline__ void async_wait0() {
#if USE_ASYNC
    __builtin_amdgcn_s_wait_asynccnt(0);
#endif
}

// ---------------------------------------------------------------------------
// float atomic max via ordered-int trick (amax initialized to -3e38f)
// ---------------------------------------------------------------------------
__device__ __forceinline__ void atomicMaxF(float* addr, float val) {
    if (val >= 0.0f)
        atomicMax((int*)addr, __float_as_int(val));
    else
        atomicMin((unsigned int*)addr, __float_as_uint(val));
}

// ---------------------------------------------------------------------------
// init: amax <- -inf-ish, accumulators <- 0 (den, vdot, ox, vagg contiguous)
// ---------------------------------------------------------------------------
__global__ void init_k(float* __restrict__ amax, long amaxN,
                       float* __restrict__ zero, long zeroN) {
    long i = (long)blockIdx.x * blockDim.x + threadIdx.x;
    if (i < amaxN)                amax[i] = -3.0e38f;
    else if (i < amaxN + zeroN)   zero[i - amaxN] = 0.0f;
}

// ---------------------------------------------------------------------------
// QKV node GEMM: z[n, 0:512] = h[n,:] @ [Wq | Wk | Wv] + [bq | bk | bv]
// block = 16 rows x 512 cols, 8 waves, wave owns 4 16-col tiles.
// A (16x128) resident in LDS; W streamed in 16-K chunks (16x512) through LDS
// via async b128 copies; K=128 -> 8 chunks, 4 v_wmma_f32_16x16x4_f32 per
// k-step per wave.
//
// WMMA f32 16x16x4 fragment layout (ISA 7.12.2):
//   A: lane<16 -> M=lane, {K=k0,k0+1}; lane>=16 -> M=lane-16, {K=k0+2,k0+3}
//   B: lane<16 -> N=lane, {K=k0,k0+1}; lane>=16 -> N=lane-16, {K=k0+2,k0+3}
//   C/D: VGPR r -> M = r + 8*(lane>=16), N = lane&15
// ---------------------------------------------------------------------------
__global__ void qkv_gemm(const float* __restrict__ h,
                         const float* __restrict__ Wq, const float* __restrict__ bq,
                         const float* __restrict__ Wk, const float* __restrict__ bk,
                         const float* __restrict__ Wv, const float* __restrict__ bv,
                         float* __restrict__ z) {
    __shared__ float ldsA[16 * LDA];
    __shared__ float ldsW[16 * LDW_QKV];
    const int tid = threadIdx.x;
    const int n0  = blockIdx.x * 16;

    // A tile: 16 rows x 128 floats = 512 float4, 2 per thread
#pragma unroll
    for (int i = 0; i < 2; ++i) {
        int f   = tid + i * 256;
        int row = f >> 5;            // 32 float4 per row
        int c4  = f & 31;
        cp_b128(h + (size_t)(n0 + row) * HID + c4 * 4, ldsA + row * LDA + c4 * 4);
    }

    // first W chunk (K rows 0..15): 16 x 512 = 2048 float4, 8 per thread
#pragma unroll
    for (int i = 0; i < 8; ++i) {
        int f   = tid + i * 256;
        int row = f >> 7;            // 128 float4 per 512-col row
        int c4  = f & 127;
        const float* src;
        if (c4 < 32)      src = Wq + (size_t)row * 128 + c4 * 4;
        else if (c4 < 64) src = Wk + (size_t)row * 128 + (c4 - 32) * 4;
        else              src = Wv + (size_t)row * 256 + (c4 - 64) * 4;
        cp_b128(src, ldsW + row * LDW_QKV + c4 * 4);
    }
    async_wait0();
    __syncthreads();

    const int wave  = tid >> 5;
    const int lane  = tid & 31;
    const int mrow  = lane & 15;
    const int khalf = (lane >> 4) << 1;   // 0 or 2
    const int ncol  = lane & 15;

    const float* bt[4]; int cb[4]; int zc[4];
#pragma unroll
    for (int t = 0; t < 4; ++t) {
        int ct = wave * 4 + t;            // 0..31 column tile
        if (ct < 8)       { bt[t] = bq; cb[t] = ct * 16; }
        else if (ct < 16) { bt[t] = bk; cb[t] = (ct - 8) * 16; }
        else              { bt[t] = bv; cb[t] = (ct - 16) * 16; }
        zc[t] = ct * 16;
    }

    v8f acc[4] = {};
    for (int kc = 0; kc < 8; ++kc) {
        const int kbase = kc * 16;
#pragma unroll
        for (int kk = 0; kk < 16; kk += 4) {
            v2f a;
            a.x = ldsA[mrow * LDA + kbase + kk + khalf];
            a.y = ldsA[mrow * LDA + kbase + kk + khalf + 1];
#pragma unroll
            for (int t = 0; t < 4; ++t) {
                v2f b;
                b.x = ldsW[(kk + khalf)     * LDW_QKV + zc[t] + ncol];
                b.y = ldsW[(kk + khalf + 1) * LDW_QKV + zc[t] + ncol];
                acc[t] = __builtin_amdgcn_wmma_f32_16x16x4_f32(
                    false, a, false, b, (short)0, acc[t], false, false);
            }
        }
        if (kc < 7) {
            __syncthreads();      // all waves done reading current chunk
            const int knext = (kc + 1) * 16;
#pragma unroll
            for (int i = 0; i < 8; ++i) {
                int f   = tid + i * 256;
                int row = f >> 7;
                int c4  = f & 127;
                const float* src;
                if (c4 < 32)      src = Wq + (size_t)(knext + row) * 128 + c4 * 4;
                else if (c4 < 64) src = Wk + (size_t)(knext + row) * 128 + (c4 - 32) * 4;
                else              src = Wv + (size_t)(knext + row) * 256 + (c4 - 64) * 4;
                cp_b128(src, ldsW + row * LDW_QKV + c4 * 4);
            }
            async_wait0();
            __syncthreads();
        }
    }

#pragma unroll
    for (int t = 0; t < 4; ++t) {
        float bias = bt[t][cb[t] + ncol];
#pragma unroll
        for (int r = 0; r < 8; ++r) {
            int row = r + ((lane >> 4) << 3);
            z[(size_t)(n0 + row) * 512 + zc[t] + ncol] = acc[t][r] + bias;
        }
    }
}

// ---------------------------------------------------------------------------
// vec GEMM: vp = vec(N*3,128) @ Wvec(128,384). 16-col tile ct -> (head=ct/3,
// field=ct%3). Wave w = head w, 3 tiles (vec1/vec2/vec3). vec3 stored;
// vec1*vec2 reduced over the 3 spatial rows via atomicAdd into vec_dot.
// ---------------------------------------------------------------------------
__global__ void vec_gemm(const float* __restrict__ vec,
                         const float* __restrict__ Wvec,
                         float* __restrict__ vec3buf,
                         float* __restrict__ vec_dot) {
    __shared__ float ldsA[16 * LDA];
    __shared__ float ldsW[16 * LDW_VEC];
    const int tid = threadIdx.x;
    const int r0  = blockIdx.x * 16;     // row index into (N*3, 128)

#pragma unroll
    for (int i = 0; i < 2; ++i) {
        int f   = tid + i * 256;
        int row = f >> 5;
        int c4  = f & 31;
        cp_b128(vec + (size_t)(r0 + row) * HID + c4 * 4, ldsA + row * LDA + c4 * 4);
    }
    // W chunk: 16 x 384 = 1536 float4, 6 per thread
#pragma unroll
    for (int i = 0; i < 6; ++i) {
        int f   = tid + i * 256;
        int row = f / 96;                // 96 float4 per 384-col row
        int c4  = f - row * 96;
        cp_b128(Wvec + (size_t)row * 384 + c4 * 4, ldsW + row * LDW_VEC + c4 * 4);
    }
    async_wait0();
    __syncthreads();

    const int head  = tid >> 5;
    const int lane  = tid & 31;
    const int mrow  = lane & 15;
    const int khalf = (lane >> 4) << 1;
    const int ncol  = lane & 15;

    v8f acc[3] = {};
    for (int kc = 0; kc < 8; ++kc) {
        const int kbase = kc * 16;
#pragma unroll
        for (int kk = 0; kk < 16; kk += 4) {
            v2f a;
            a.x = ldsA[mrow * LDA + kbase + kk + khalf];
            a.y = ldsA[mrow * LDA + kbase + kk + khalf + 1];
#pragma unroll
            for (int f = 0; f < 3; ++f) {
                int col = head * 48 + f * 16 + ncol;
                v2f b;
                b.x = ldsW[(kk + khalf)     * LDW_VEC + col];
                b.y = ldsW[(kk + khalf + 1) * LDW_VEC + col];
                acc[f] = __builtin_amdgcn_wmma_f32_16x16x4_f32(
                    false, a, false, b, (short)0, acc[f], false, false);
            }
        }
        if (kc < 7) {
            __syncthreads();
            const int knext = (kc + 1) * 16;
#pragma unroll
            for (int i = 0; i < 6; ++i) {
                int f   = tid + i * 256;
                int row = f / 96;
                int c4  = f - row * 96;
                cp_b128(Wvec + (size_t)(knext + row) * 384 + c4 * 4,
                        ldsW + row * LDW_VEC + c4 * 4);
            }
            async_wait0();
            __syncthreads();
        }
    }

#pragma unroll
    for (int r = 0; r < 8; ++r) {
        int row  = r + ((lane >> 4) << 3);
        int grow = r0 + row;                 // (node*3 + d)
        vec3buf[(size_t)grow * HID + head * 16 + ncol] = acc[2][r];
        float p  = acc[0][r] * acc[1][r];
        int node = grow / 3;
        atomicAdd(&vec_dot[(size_t)node * HID + head * 16 + ncol], p);
    }
}

// ---------------------------------------------------------------------------
// Pass A: alpha_ij = (q[dst,h,:] . k[src,h,:]) / 4 ; segment atomic-max
// ---------------------------------------------------------------------------
__global__ void edge_alpha(const int* __restrict__ ei, const float* __restrict__ z,
                           float* __restrict__ alphabuf, float* __restrict__ amax,
                           int E) {
    int t = blockIdx.x * blockDim.x + threadIdx.x;
    if (t >= E * NHEAD) return;
    int e = t >> 3, hd = t & 7;
    int src = ei[e], dst = ei[E + e];
    const float4* q = (const float4*)(z + (size_t)dst * 512 + hd * 16);
    const float4* k = (const float4*)(z + (size_t)src * 512 + 128 + hd * 16);
    float a = 0.0f;
#pragma unroll
    for (int i = 0; i < 4; ++i) {
        float4 qv = q[i], kv = k[i];
        a += qv.x * kv.x + qv.y * kv.y + qv.z * kv.z + qv.w * kv.w;
    }
    a *= 0.25f;  // 1/sqrt(C), C=16
    alphabuf[t] = a;
    atomicMaxF(&amax[(size_t)dst * NHEAD + hd], a);
}

// ---------------------------------------------------------------------------
// Pass B: ex = exp(alpha - amax[dst]); segment atomic-sum of ex
// ---------------------------------------------------------------------------
__global__ void edge_exp(const int* __restrict__ ei, float* __restrict__ exb,
                         const float* __restrict__ amax, float* __restrict__ den,
                         int E) {
    int t = blockIdx.x * blockDim.x + threadIdx.x;
    if (t >= E * NHEAD) return;
    int e = t >> 3, hd = t & 7;
    int dst = ei[E + e];
    float ex = __expf(exb[t] - amax[(size_t)dst * NHEAD + hd]);
    exb[t] = ex;
    atomicAdd(&den[(size_t)dst * NHEAD + hd], ex);
}

// ---------------------------------------------------------------------------
// Pass C: normalized scatter. one thread per (edge, hc).
// ---------------------------------------------------------------------------
__global__ void edge_scatter(const int* __restrict__ ei, const float* __restrict__ z,
                             const float* __restrict__ vec,
                             const float* __restrict__ exb, const float* __restrict__ den,
                             float* __restrict__ ox, float* __restrict__ vagg, int E) {
    long t = (long)blockIdx.x * blockDim.x + threadIdx.x;
    if (t >= (long)E * HID) return;
    int e  = (int)(t >> 7);
    int hc = (int)(t & 127);
    int hd = hc >> 4, c = hc & 15;
    int src = ei[e], dst = ei[E + e];
    float w = exb[(size_t)e * NHEAD + hd] / den[(size_t)dst * NHEAD + hd];
    const float* vrow = z + (size_t)src * 512 + 256 + hd * 32;
    float hm = vrow[c];
    float s1 = vrow[16 + c];
    atomicAdd(&ox[(size_t)dst * HID + hc], hm * w);
    const float* vs = vec  + (size_t)src * 3 * HID;
    float*       vd = vagg + (size_t)dst * 3 * HID;
#pragma unroll
    for (int d = 0; d < 3; ++d)
        atomicAdd(&vd[d * HID + hc], vs[d * HID + hc] * s1);
}

// ---------------------------------------------------------------------------
// final: qp = out_x(n,h,:) @ Wp(16,48) + bp -> q1,q2,q3
//   h_out = h + q2 + q3*vec_dot ; vec_out = vec + vec3*q1 + vec_agg
// ---------------------------------------------------------------------------
__global__ void node_final(const float* __restrict__ h, const float* __restrict__ vec,
                           const float* __restrict__ ox, const float* __restrict__ vdot,
                           const float* __restrict__ v3, const float* __restrict__ vagg,
                           const float* __restrict__ Wp, const float* __restrict__ bp,
                           float* __restrict__ out, int N) {
    long t = (long)blockIdx.x * blockDim.x + threadIdx.x;
    if (t >= (long)N * HID) return;
    int n  = (int)(t >> 7);
    int hc = (int)(t & 127);
    int hd = hc >> 4, c = hc & 15;
    const float* oxr = ox + (size_t)n * HID + hd * 16;
    float q1 = bp[c], q2 = bp[16 + c], q3 = bp[32 + c];
#pragma unroll
    for (int k = 0; k < 16; ++k) {
        float o = oxr[k];
        const float* wr = Wp + k * 48;
        q1 += o * wr[c];
        q2 += o * wr[16 + c];
        q3 += o * wr[32 + c];
    }
    out[t] = h[t] + q2 + q3 * vdot[t];
    size_t HOFF = (size_t)N * HID;
#pragma unroll
    for (int d = 0; d < 3; ++d) {
        size_t i = ((size_t)n * 3 + d) * HID + hc;
        out[HOFF + i] = vec[i] + v3[i] * q1 + vagg[i];
    }
}

// ---------------------------------------------------------------------------
extern "C" void kernel_launch(void* const* d_in, const int* in_sizes, int n_in,
                              void* d_out, int out_size, void* d_ws, size_t ws_size,
                              hipStream_t stream) {
    const float* h    = (const float*)d_in[0];
    const float* vec  = (const float*)d_in[1];
    const int*   ei   = (const int*)  d_in[2];
    const float* Wq   = (const float*)d_in[3];
    const float* bq   = (const float*)d_in[4];
    const float* Wk   = (const float*)d_in[5];
    const float* bk   = (const float*)d_in[6];
    const float* Wv   = (const float*)d_in[7];
    const float* bv   = (const float*)d_in[8];
    const float* Wp   = (const float*)d_in[9];
    const float* bp   = (const float*)d_in[10];
    const float* Wvec = (const float*)d_in[11];

    const int N = in_sizes[0] / HID;   // 20000
    const int E = in_sizes[2] / 2;     // 640000

    float* ws = (float*)d_ws;
    size_t off = 0;
    float* zbuf = ws + off; off += (size_t)N * 512;   // QKV
    float* v3   = ws + off; off += (size_t)N * 384;   // vec3
    float* exb  = ws + off; off += (size_t)E * NHEAD; // alpha -> ex
    float* amax = ws + off; off += (size_t)N * NHEAD;
    float* den  = ws + off; off += (size_t)N * NHEAD; // start of zeroed region
    float* vdot = ws + off; off += (size_t)N * HID;
    float* oxb  = ws + off; off += (size_t)N * HID;
    float* vagg = ws + off; off += (size_t)N * 384;

    long amaxN = (long)N * NHEAD;
    long zeroN = (long)N * NHEAD + (long)N * HID * 2 + (long)N * 384;
    long initN = amaxN + zeroN;

    init_k<<<(unsigned)((initN + 255) / 256), 256, 0, stream>>>(amax, amaxN, den, zeroN);
    qkv_gemm<<<(unsigned)(N / 16), 256, 0, stream>>>(h, Wq, bq, Wk, bk, Wv, bv, zbuf);
    vec_gemm<<<(unsigned)((3 * N) / 16), 256, 0, stream>>>(vec, Wvec, v3, vdot);
    edge_alpha<<<(unsigned)(((long)E * NHEAD + 255) / 256), 256, 0, stream>>>(ei, zbuf, exb, amax, E);
    edge_exp<<<(unsigned)(((long)E * NHEAD + 255) / 256), 256, 0, stream>>>(ei, exb, amax, den, E);
    edge_scatter<<<(unsigned)(((long)E * HID + 255) / 256), 256, 0, stream>>>(ei, zbuf, vec, exb, den, oxb, vagg, E);
    node_final<<<(unsigned)(((long)N * HID + 255) / 256), 256, 0, stream>>>(h, vec, oxb, vdot, v3, vagg, Wp, bp, (float*)d_out, N);
}